// Spiking_LayerNorm_3092376453491
// MI455X (gfx1250) — compile-verified
//
#include <hip/hip_runtime.h>
#include <hip/hip_bf16.h>

// Spiking LayerNorm, T=4, D=768, rows = 64*196 per time step.
// Bandwidth-bound (~308 MB @ 23.3 TB/s ~= 13 us). Strategy:
//   - one wave32 per row; X / Y_pre / gamma / beta live in VGPRs across all 4 steps
//   - global -> LDS async DMA (gfx1250 GLOBAL_LOAD_ASYNC_TO_LDS_B128, ASYNCcnt)
//     double-buffered to prefetch time step t+2 while computing step t
//   - non-temporal 128-bit stores for the write-once output stream

typedef float v4f __attribute__((ext_vector_type(4)));

#define TT 4
#define DD 768
#define NROWS (64 * 196)        // rows per time step = 12544
#define WPB 8                   // waves per block
#define CHUNKS 6                // 6 * (32 lanes * 4 floats) = 768 floats per row

__global__ __launch_bounds__(256) void spiking_ln_kernel(
    const float* __restrict__ in, const float* __restrict__ weight,
    const float* __restrict__ bias, float* __restrict__ out) {
  // per block: 8 waves * 2 buffers * 192 float4 (3 KB each) = 48 KB LDS
  __shared__ v4f sbuf[WPB * 2 * 192];

  const int lane = threadIdx.x & 31;
  const int wv   = threadIdx.x >> 5;
  const int row  = blockIdx.x * WPB + wv;
  if (row >= NROWS) return;   // grid divides exactly; uniform per wave anyway

  const size_t rbase = (size_t)row * DD;     // float offset of this row in one step
  const size_t tstep = (size_t)NROWS * DD;   // float stride between time steps

  v4f* const buf0 = &sbuf[wv * 2 * 192];
  v4f* const buf1 = buf0 + 192;

  // Issue one row (6 x b128 per lane-batch) as async global->LDS DMA.
  auto issue = [&](const float* gsrc, v4f* lbuf) {
#pragma unroll
    for (int c = 0; c < CHUNKS; ++c) {
      // AS3 offset = low 32 bits of the generic LDS address
      unsigned ldsa = (unsigned)(uintptr_t)(&lbuf[c * 32 + lane]);
      unsigned long long ga =
          (unsigned long long)(uintptr_t)(gsrc + c * 128 + lane * 4);
      asm volatile("global_load_async_to_lds_b128 %0, %1, off"
                   :: "v"(ldsa), "v"(ga) : "memory");
    }
  };

  // gamma / beta: load once, keep in VGPRs (tiny, L2-resident)
  v4f wr[CHUNKS], br[CHUNKS];
#pragma unroll
  for (int c = 0; c < CHUNKS; ++c) {
    wr[c] = *(const v4f*)(weight + c * 128 + lane * 4);
    br[c] = *(const v4f*)(bias   + c * 128 + lane * 4);
  }

  const v4f vzero = {0.f, 0.f, 0.f, 0.f};
  v4f X[CHUNKS], Yp[CHUNKS];
#pragma unroll
  for (int c = 0; c < CHUNKS; ++c) { X[c] = vzero; Yp[c] = vzero; }

  // accumulate membrane state from an LDS buffer
  auto accum = [&](v4f* lbuf) {
#pragma unroll
    for (int c = 0; c < CHUNKS; ++c) X[c] += lbuf[c * 32 + lane];
  };

  // layernorm * scale, temporal difference, NT store
  auto stats_store = [&](int t) {
    float s1 = 0.f, s2 = 0.f;
#pragma unroll
    for (int c = 0; c < CHUNKS; ++c) {
#pragma unroll
      for (int k = 0; k < 4; ++k) { float v = X[c][k]; s1 += v; s2 += v * v; }
    }
#pragma unroll
    for (int off = 16; off > 0; off >>= 1) {
      s1 += __shfl_xor(s1, off, 32);
      s2 += __shfl_xor(s2, off, 32);
    }
    const float inv_d = 1.0f / (float)DD;
    const float mean  = s1 * inv_d;
    const float var   = s2 * inv_d - mean * mean;
    const float rstd  = rsqrtf(var + 1e-5f);
    const float scale = (float)(t + 1) * 0.25f;
    const float a     = rstd * scale;
    float* op = out + (size_t)t * tstep + rbase;
#pragma unroll
    for (int c = 0; c < CHUNKS; ++c) {
      v4f Y = (X[c] - mean) * a * wr[c] + br[c] * scale;
      v4f o = Y - Yp[c];
      Yp[c] = Y;
      __builtin_nontemporal_store(o, (v4f*)(op + c * 128 + lane * 4));
    }
  };

  // prime the pipeline: batches for t=0 and t=1 (ASYNCcnt = 12)
  issue(in + rbase,         buf0);
  issue(in + rbase + tstep, buf1);

  // t = 0 : oldest 6 done -> buf0 ready
  asm volatile("s_wait_asynccnt 0x6" ::: "memory");
  accum(buf0);
  asm volatile("s_wait_dscnt 0x0" ::: "memory");   // retire LDS reads (WAR)
  issue(in + rbase + 2 * tstep, buf0);             // prefetch t=2
  stats_store(0);

  // t = 1
  asm volatile("s_wait_asynccnt 0x6" ::: "memory");
  accum(buf1);
  asm volatile("s_wait_dscnt 0x0" ::: "memory");
  issue(in + rbase + 3 * tstep, buf1);             // prefetch t=3
  stats_store(1);

  // t = 2
  asm volatile("s_wait_asynccnt 0x6" ::: "memory");
  accum(buf0);
  stats_store(2);

  // t = 3
  asm volatile("s_wait_asynccnt 0x0" ::: "memory");
  accum(buf1);
  stats_store(3);
}

extern "C" void kernel_launch(void* const* d_in, const int* in_sizes, int n_in,
                              void* d_out, int out_size, void* d_ws, size_t ws_size,
                              hipStream_t stream) {
  const float* in  = (const float*)d_in[0];
  const float* w   = (const float*)d_in[1];
  const float* b   = (const float*)d_in[2];
  float* out       = (float*)d_out;

  dim3 grid(NROWS / WPB);   // 1568 blocks
  dim3 block(256);          // 8 wave32 per block
  spiking_ln_kernel<<<grid, block, 0, stream>>>(in, w, b, out);
}